// seasonality_extraction_87986700026365
// MI455X (gfx1250) — compile-verified
//
#include <hip/hip_runtime.h>

typedef __attribute__((ext_vector_type(4))) float v4f;

// Problem constants (B, L, E, T from the reference)
constexpr int Bc    = 16;
constexpr int Lc    = 192;
constexpr int Ec    = 128;
constexpr int Tc    = 25;
constexpr int NI    = Lc - Tc - 1;          // 166 shifted rows
constexpr int OUTL  = (Tc + 1) + NI * Lc;   // 26 + 31872 = 31898 output rows per batch
constexpr int E4    = Ec / 4;               // 32 float4 per row
constexpr int MAINB = Bc * NI;              // 2656 main blocks
constexpr int ROWS_PER_PHASE = 96;          // 48 KB LDS per phase
constexpr int PHASE_V4 = ROWS_PER_PHASE * E4; // 3072 float4
constexpr int STAGE_ITERS = PHASE_V4 / 256;   // 12 async b128 per thread per phase

// ---------------------------------------------------------------------------
// Kernel 1: s0 = softmax(x*W+b)[...,0] = 1 / (1 + exp(x*(W1-W0) + (b1-b0)))
// 393216 elements -> 98304 float4; trivial (~0.15 us), keeps v_exp out of the
// bandwidth-bound main loop. v_rcp_f32 instead of full IEEE divide.
// ---------------------------------------------------------------------------
__global__ __launch_bounds__(256) void gate_kernel(const float* __restrict__ x,
                                                   const float* __restrict__ W,
                                                   const float* __restrict__ bvec,
                                                   float* __restrict__ s0) {
    int n = blockIdx.x * 256 + threadIdx.x;          // float4 index
    float dw = W[1] - W[0];
    float db = bvec[1] - bvec[0];
    v4f xv = ((const v4f*)x)[n];
    v4f s;
    s.x = __builtin_amdgcn_rcpf(1.0f + __expf(xv.x * dw + db));
    s.y = __builtin_amdgcn_rcpf(1.0f + __expf(xv.y * dw + db));
    s.z = __builtin_amdgcn_rcpf(1.0f + __expf(xv.z * dw + db));
    s.w = __builtin_amdgcn_rcpf(1.0f + __expf(xv.w * dw + db));
    ((v4f*)s0)[n] = s;
}

// ---------------------------------------------------------------------------
// Kernel 2: one block per (b, i). Stage s0[b] slab into LDS via CDNA5 async
// copies (ASYNCcnt-tracked), then stream 192 output rows with NT b128 stores.
// out[b, 26 + i*192 + l, e] = xp + (xi - xp) * s0[b, l, e]
//   xi = x[b, 26+i, e], xp = x[b, 1+i, e]
// Blocks >= MAINB copy the head rows x[:, :26, :].
// ---------------------------------------------------------------------------
__global__ __launch_bounds__(256) void blocks_kernel(const float* __restrict__ xg,
                                                     const float* __restrict__ s0g,
                                                     float* __restrict__ outg) {
    __shared__ v4f sbuf[PHASE_V4];                   // 48 KB
    const v4f* x4   = (const v4f*)xg;
    const v4f* s4   = (const v4f*)s0g;
    v4f*       out4 = (v4f*)outg;
    const int bid = blockIdx.x;
    const int tid = threadIdx.x;

    if (bid < MAINB) {
        const int b  = bid / NI;
        const int i  = bid - b * NI;
        const int e4 = tid & (E4 - 1);               // 0..31: one wave = one 512B row segment
        const int lo = tid >> 5;                     // 0..7:  8 rows per iteration

        // Per-thread invariant row data (L2-resident, 512 B per row)
        v4f xi = x4[(b * Lc + (Tc + 1) + i) * E4 + e4];
        v4f xp = x4[(b * Lc + 1 + i) * E4 + e4];
        v4f d  = xi - xp;                            // out = fma(s0, d, xp)

        const long Rbase = (long)b * OUTL + (Tc + 1) + (long)i * Lc;

        for (int p = 0; p < 2; ++p) {
            if (p) __syncthreads();                  // phase-0 reads done before overwrite

            // Async stage 96 rows of s0[b] (48 KB) into LDS: 12 back-to-back
            // per-lane 16 B async copies (512 B per wave instruction), then a
            // single s_wait_asynccnt 0.
            const v4f* src = s4 + ((long)b * Lc + p * ROWS_PER_PHASE) * E4;
#pragma unroll
            for (int j = 0; j < STAGE_ITERS; ++j) {
                const int k = tid + j * 256;
                unsigned loff = (unsigned)(unsigned long long)&sbuf[k]; // LDS byte offset
                const v4f* g = src + k;
                asm volatile("global_load_async_to_lds_b128 %0, %1, off"
                             :: "v"(loff), "v"(g) : "memory");
            }
            asm volatile("s_wait_asynccnt 0" ::: "memory");
            __syncthreads();

            // Stream 96 rows: 12 iters x (1 ds_load_b128 + 4 v_fma + 1 NT store_b128)
#pragma unroll
            for (int l8 = 0; l8 < ROWS_PER_PHASE; l8 += 8) {
                const int r = l8 + lo;
                v4f sv = sbuf[r * E4 + e4];
                v4f o  = sv * d + xp;
                __builtin_nontemporal_store(
                    o, &out4[(Rbase + p * ROWS_PER_PHASE + r) * E4 + e4]);
            }
        }
    } else {
        // Head copy: out[:, :26, :] = x[:, :26, :]  (13312 float4 over 52 blocks)
        const int HEAD_B = (Tc + 1) * E4;            // 832 float4 per batch
        int n = (bid - MAINB) * 256 + tid;
        if (n < Bc * HEAD_B) {
            int b   = n / HEAD_B;
            int rem = n - b * HEAD_B;
            v4f v = x4[(long)b * Lc * E4 + rem];
            __builtin_nontemporal_store(v, &out4[(long)b * OUTL * E4 + rem]);
        }
    }
}

// ---------------------------------------------------------------------------
extern "C" void kernel_launch(void* const* d_in, const int* in_sizes, int n_in,
                              void* d_out, int out_size, void* d_ws, size_t ws_size,
                              hipStream_t stream) {
    (void)in_sizes; (void)n_in; (void)out_size; (void)ws_size;
    const float* x    = (const float*)d_in[0];
    const float* W    = (const float*)d_in[1];
    const float* bvec = (const float*)d_in[2];
    float*       out  = (float*)d_out;
    float*       s0   = (float*)d_ws;                // needs B*L*E*4 = 1.5 MB

    // Kernel 1: 98304 float4 -> 384 blocks x 256 threads
    gate_kernel<<<384, 256, 0, stream>>>(x, W, bvec, s0);

    // Kernel 2: 2656 main blocks + 52 head-copy blocks
    blocks_kernel<<<MAINB + 52, 256, 0, stream>>>(x, s0, out);
}